// Microscope_51762945851926
// MI455X (gfx1250) — compile-verified
//
#include <hip/hip_runtime.h>
#include <hip/hip_bf16.h>

// Problem constants (match reference)
#define BS     4
#define CH     1
#define HDIM   256
#define WDIM   256
#define DDIM   64
#define PH     21
#define PATCH  (PH * PH * PH)      // 9261
#define N_EMIT 10000
#define OUTSCALE 1000000.0f        // SCALE * MULT = 1e6

#define NTHREADS 256
#define NITER    ((PATCH + NTHREADS - 1) / NTHREADS)   // 37
#define NFULL    (PATCH / NTHREADS)                    // 36 full iterations
#define TAIL     (PATCH - NFULL * NTHREADS)            // 45 threads do iter 36

typedef __attribute__((ext_vector_type(2))) float v2f;
typedef __attribute__((ext_vector_type(8))) float v8f;

// ---------------------------------------------------------------------------
// Zero the output volume (harness poisons d_out; atomics need a clean base).
// ---------------------------------------------------------------------------
__global__ void zero_out_kernel(float4* __restrict__ out, int n4) {
    int i = blockIdx.x * blockDim.x + threadIdx.x;
    if (i < n4) out[i] = make_float4(0.f, 0.f, 0.f, 0.f);
}

// ---------------------------------------------------------------------------
// Wave32 sum reduction using V_WMMA_F32_16X16X4_F32.
// A (16x4 f32) = 2 VGPRs/lane: lane partial in slot 0, zero in slot 1 ->
// all 32 lane partials occupy the A tile. B = all ones, C = 0.
// D[m][n] = sum_k A[m,k]; summing a lane's 8 D registers gives the column
// sum over its 8 rows; shfl_xor(16) folds the two row-halves -> every lane
// holds the full wave total.  (Executed with EXEC all-ones: no divergence.)
// ---------------------------------------------------------------------------
__device__ __forceinline__ float wave_sum_wmma(float x) {
    v2f a; a[0] = x;    a[1] = 0.0f;
    v2f b; b[0] = 1.0f; b[1] = 1.0f;
    v8f c = {};
    v8f d = __builtin_amdgcn_wmma_f32_16x16x4_f32(
        /*neg_a=*/false, a, /*neg_b=*/false, b,
        /*c_mod=*/(short)0, c, /*reuse_a=*/false, /*reuse_b=*/false);
    float p = d[0] + d[1] + d[2] + d[3] + d[4] + d[5] + d[6] + d[7];
    p += __shfl_xor(p, 16, 32);   // combine M=0..7 half with M=8..15 half
    return p;                      // every lane: total over all 32 lanes
}

// ---------------------------------------------------------------------------
// One workgroup (256 threads = 8 wave32) per emitter. Patch values live in
// registers (each thread re-reads only its own elements, so no LDS staging
// is needed):
//   1) stream patch HBM -> registers with clamp, per-thread partial sum
//   2) WMMA wave reduction + tiny LDS combine -> emitter sum
//   3) scatter registers * (i_val * 1e6 / sum) via global_atomic_add_f32,
//      bounds check == the PAD crop in the reference (PAD=12 >= PH/2=10)
// ---------------------------------------------------------------------------
__global__ void __launch_bounds__(NTHREADS)
place_psf_kernel(const float* __restrict__ psf_raw,
                 const float* __restrict__ i_val,
                 const int*   __restrict__ bb,
                 const int*   __restrict__ cc,
                 const int*   __restrict__ hh,
                 const int*   __restrict__ ww,
                 const int*   __restrict__ dd,
                 float*       __restrict__ out) {
    __shared__ float wave_sums[8];

    const int e   = blockIdx.x;
    const int tid = threadIdx.x;
    const int wid = tid >> 5;

    const float* __restrict__ src = psf_raw + (size_t)e * PATCH;

    // Phase 1: load + clamp into registers, per-thread partial sum.
    // Elements i = tid + 256k, k = 0..36 (k==36 only for tid < 45).
    float v[NITER];
    float part = 0.0f;
    #pragma unroll
    for (int k = 0; k < NITER; ++k) {
        float x = 0.0f;
        if (k < NFULL || tid < TAIL)
            x = fmaxf(src[tid + NTHREADS * k], 0.0f);
        v[k] = x;
        part += x;
    }

    // Phase 2: block reduction (all 256 threads converged here)
    float wtot = wave_sum_wmma(part);
    if ((tid & 31) == 0) wave_sums[wid] = wtot;
    __syncthreads();
    float total = wave_sums[0] + wave_sums[1] + wave_sums[2] + wave_sums[3]
                + wave_sums[4] + wave_sums[5] + wave_sums[6] + wave_sums[7];

    // Per-emitter scale: i_val * 1e6 / sum
    const float scale = i_val[e] * (OUTSCALE / total);

    const int b  = bb[e];
    const int c  = cc[e];                 // CH == 1, c == 0 always
    const int h0 = hh[e] - PH / 2;
    const int w0 = ww[e] - PH / 2;
    const int d0 = dd[e] - PH / 2;
    const int bc = b * CH + c;

    // Phase 3: scatter with crop. Incremental base-21 digit stepping:
    // i += 256  <=>  pd += 4, pw += 12 (256 = 12*21 + 4), single carries.
    int pd = tid % PH;
    int t  = tid / PH;
    int pw = t % PH;
    int ph = t / PH;

    #pragma unroll
    for (int k = 0; k < NITER; ++k) {
        if (k < NFULL || tid < TAIL) {
            int zh = h0 + ph;
            int zw = w0 + pw;
            int zd = d0 + pd;
            if ((unsigned)zh < HDIM && (unsigned)zw < WDIM && (unsigned)zd < DDIM) {
                size_t oi = (((size_t)bc * HDIM + zh) * WDIM + zw) * DDIM + zd;
                atomicAdd(&out[oi], v[k] * scale);
            }
        }
        // advance (pd,pw,ph) by 256 in base-21
        pd += 4;  if (pd >= PH) { pd -= PH; ++pw; }
        pw += 12; if (pw >= PH) { pw -= PH; ++ph; }
    }
}

extern "C" void kernel_launch(void* const* d_in, const int* in_sizes, int n_in,
                              void* d_out, int out_size, void* d_ws, size_t ws_size,
                              hipStream_t stream) {
    const float* psf_raw = (const float*)d_in[0];
    const float* i_val   = (const float*)d_in[1];
    const int*   b       = (const int*)d_in[2];
    const int*   c       = (const int*)d_in[3];
    const int*   h       = (const int*)d_in[4];
    const int*   w       = (const int*)d_in[5];
    const int*   d       = (const int*)d_in[6];
    float*       out     = (float*)d_out;

    // 1) clear output (poisoned by harness; atomics accumulate on top)
    int n4 = out_size / 4;
    int zb = (n4 + NTHREADS - 1) / NTHREADS;
    zero_out_kernel<<<zb, NTHREADS, 0, stream>>>((float4*)out, n4);

    // 2) fused normalize + scatter, one workgroup per emitter
    place_psf_kernel<<<N_EMIT, NTHREADS, 0, stream>>>(psf_raw, i_val, b, c, h, w, d, out);
}